// CompressedInteractionNetwork_34050500723221
// MI455X (gfx1250) — compile-verified
//
#include <hip/hip_runtime.h>

// CIN (xDeepFM) forward on gfx1250 using V_WMMA_F32_16X16X4_F32.
// B=1024, m=32, d=64, H=256. Columns c = b*64 + d (N = 65536).
// Per layer:  D[r,c] = sum_m x0[m,c] * (W_m^T h)[r,c], relu(+bias),
// split into fc-contribution rows and next-layer h rows.

typedef float v2f __attribute__((ext_vector_type(2)));
typedef float v8f __attribute__((ext_vector_type(8)));

#define NFIELD 32
#define HID    256
#define NCOLS  65536      // 1024 batches * 64 embed dims
#define CT     4          // 4 col-tiles of 16 = 64 columns = one batch element

__global__ void cin_init_out(float* __restrict__ out, const float* __restrict__ fcb) {
  int i = blockIdx.x * blockDim.x + threadIdx.x;
  if (i < 1024) out[i] = fcb[0];
}

template <int NPREV, bool HSRC_IS_X, bool LAST>
__global__ __launch_bounds__(512) void cin_layer(
    const float* __restrict__ x,     // (1024, 32, 64)
    const float* __restrict__ hsrc,  // [NPREV][NCOLS] (ignored if HSRC_IS_X)
    const float* __restrict__ W,     // (32*NPREV, 256)
    const float* __restrict__ bias,  // (256)
    const float* __restrict__ fcW,   // (512)
    int fc_base,                     // 0 / 128 / 256
    float* __restrict__ hdst,        // [128][NCOLS] (unused when LAST)
    float* __restrict__ out)         // (1024)
{
  constexpr int HP = NPREV + 4;                 // padded LDS stride (bank spread)
  __shared__ float x0s[NFIELD * 64];            // [m][d]  (= x[b] tile)
  __shared__ float hT[64 * HP];                 // [d][n]  transposed h tile
  __shared__ float fc_acc;

  const int tid  = threadIdx.x;
  const int lane = tid & 31;
  const int wid  = tid >> 5;          // 0..15 : which 16-row H-tile
  const int b    = blockIdx.x;        // batch element; cols c = b*64 + d
  const int mrow = lane & 15;         // row (A) or column (B/C/D) within tile
  const int koff = (lane >> 4) << 1;  // K offset 0 / 2 for A,B fragments
  const int roff = (lane >> 4) << 3;  // row offset 0 / 8 for C/D fragments

  if (tid == 0) fc_acc = 0.0f;

  // ---- stage x0 tile and transposed h tile into LDS ----
  const float* xb = x + (size_t)b * (NFIELD * 64);
  for (int i = tid; i < NFIELD * 64; i += 512) x0s[i] = xb[i];
  for (int i = tid; i < NPREV * 64; i += 512) {
    int d = i & 63, n = i >> 6;
    float hv;
    if constexpr (HSRC_IS_X) hv = xb[n * 64 + d];
    else                     hv = hsrc[(size_t)n * NCOLS + (size_t)b * 64 + d];
    hT[d * HP + n] = hv;
  }
  __syncthreads();

  const int m_base = wid * 16;
  const float* Wc = W + m_base + mrow;          // one W column per lane

  v8f acc[CT];
#pragma unroll
  for (int t = 0; t < CT; ++t) acc[t] = {0.f,0.f,0.f,0.f,0.f,0.f,0.f,0.f};

  for (int mb = 0; mb < NFIELD; ++mb) {
    v8f G[CT];
#pragma unroll
    for (int t = 0; t < CT; ++t) G[t] = {0.f,0.f,0.f,0.f,0.f,0.f,0.f,0.f};

#pragma unroll 4
    for (int kk = 0; kk < NPREV; kk += 4) {
      // A fragment: A[r,k] = W[mb*NPREV + kk + k, m_base + r]
      const int kg = mb * NPREV + kk + koff;
      v2f a;
      a.x = Wc[(size_t)kg * HID];
      a.y = Wc[(size_t)(kg + 1) * HID];
#pragma unroll
      for (int t = 0; t < CT; ++t) {
        // B fragment: B[k,c] = h[kk+k, c]; ds_load_b64 grabs rows
        // {kk,kk+1} (lanes 0-15) / {kk+2,kk+3} (lanes 16-31).
        const float2 hv = *(const float2*)&hT[(t * 16 + mrow) * HP + kk + koff];
        v2f bf; bf.x = hv.x; bf.y = hv.y;
        G[t] = __builtin_amdgcn_wmma_f32_16x16x4_f32(
            false, a, false, bf, (short)0, G[t], false, false);
      }
    }
    // D += x0[mb, c] * G   (per-column scalar == per-lane scalar)
#pragma unroll
    for (int t = 0; t < CT; ++t) {
      const float s = x0s[mb * 64 + t * 16 + mrow];
#pragma unroll
      for (int j = 0; j < 8; ++j) acc[t][j] = fmaf(s, G[t][j], acc[t][j]);
    }
  }

  // ---- epilogue: bias + relu, then fc reduction / h store ----
#pragma unroll
  for (int t = 0; t < CT; ++t) {
    const int d = t * 16 + mrow;
#pragma unroll
    for (int j = 0; j < 8; ++j) {
      const int row = m_base + j + roff;        // H row 0..255
      float v = acc[t][j] + bias[row];
      v = v > 0.0f ? v : 0.0f;
      if (LAST || wid < 8) {
        // fc path: reduce over the 16 columns (same batch b) of this tile
        float r = v;
        r += __shfl_xor(r, 1, 16);
        r += __shfl_xor(r, 2, 16);
        r += __shfl_xor(r, 4, 16);
        r += __shfl_xor(r, 8, 16);
        if (mrow == 0) atomicAdd(&fc_acc, r * fcW[fc_base + row]);
      } else {
        // h path: rows 128..255 become next layer's h[n][c]
        hdst[(size_t)(row - 128) * NCOLS + (size_t)b * 64 + d] = v;
      }
    }
  }
  __syncthreads();
  if (tid == 0) atomicAdd(out + b, fc_acc);
}

extern "C" void kernel_launch(void* const* d_in, const int* in_sizes, int n_in,
                              void* d_out, int out_size, void* d_ws, size_t ws_size,
                              hipStream_t stream) {
  (void)in_sizes; (void)n_in; (void)out_size; (void)ws_size;
  const float* x   = (const float*)d_in[0];
  const float* W0  = (const float*)d_in[1];
  const float* b0  = (const float*)d_in[2];
  const float* W1  = (const float*)d_in[3];
  const float* b1  = (const float*)d_in[4];
  const float* W2  = (const float*)d_in[5];
  const float* b2  = (const float*)d_in[6];
  const float* fcW = (const float*)d_in[7];
  const float* fcb = (const float*)d_in[8];
  float* out = (float*)d_out;

  float* h0 = (float*)d_ws;                      // 128 * 65536 floats = 32 MB
  float* h1 = h0 + (size_t)128 * NCOLS;          // +32 MB

  cin_init_out<<<4, 256, 0, stream>>>(out, fcb);
  cin_layer<32,  true,  false><<<1024, 512, 0, stream>>>(x, x,  W0, b0, fcW, 0,   h0, out);
  cin_layer<128, false, false><<<1024, 512, 0, stream>>>(x, h0, W1, b1, fcW, 128, h1, out);
  cin_layer<128, false, true ><<<1024, 512, 0, stream>>>(x, h1, W2, b2, fcW, 256, out /*unused*/, out);
}